// QuantumKernelMethod_65481071406685
// MI455X (gfx1250) — compile-verified
//
#include <hip/hip_runtime.h>
#include <math.h>

// 4-qubit quantum kernel method, restructured as:
//   out[n] = | <Z0> of  V * RYten(-y_n) * U * s(x_n) |
// U,V are batch-uniform 16x16 complex unitaries (built once per launch);
// per-element 16x16 matvecs are batched 16-wide into fp32 WMMA GEMMs
// (V_WMMA_F32_16X16X4_F32 chained over K in 4 chunks of 4).

typedef __attribute__((ext_vector_type(2))) float v2f;
typedef __attribute__((ext_vector_type(4))) float v4f;
typedef __attribute__((ext_vector_type(8))) float v8f;

#define WMMA_F32_4(A, B, C) \
    __builtin_amdgcn_wmma_f32_16x16x4_f32(false, (A), false, (B), (short)0, (C), false, false)

#define TILES_PER_WAVE 4

// ---------------------------------------------------------------------------
// Setup: build U (forward layers, +params) and V (inverse layers, -params).
// One wave: lanes 0-15 compute columns of U, lanes 16-31 columns of V.
// ws layout (floats): Ure[16][16] | Uim[16][16] | Vre[16][16] | Vim[16][16]
// Matrices stored row-major: X[m*16 + col]. Precise sincos here (runs once).
// ---------------------------------------------------------------------------
__global__ void qkm_build_unitaries(const float* __restrict__ params,
                                    float* __restrict__ ws) {
    const int tid  = threadIdx.x;
    const int col  = tid & 15;
    const float sg = (tid < 16) ? 1.0f : -1.0f;

    float c[4], s[4];
#pragma unroll
    for (int i = 0; i < 4; ++i) {
        float a = 0.5f * sg * params[i];
        c[i] = cosf(a);
        s[i] = sinf(a);
    }

    float re[16], im[16];
#pragma unroll
    for (int m = 0; m < 16; ++m) { re[m] = 0.0f; im[m] = 0.0f; }
    re[col] = 1.0f;

#pragma unroll
    for (int L = 0; L < 2; ++L) {
        // RX on wires 0..3 : RX = [[c, -i s], [-i s, c]]
#pragma unroll
        for (int w = 0; w < 4; ++w) {
            const int d = 8 >> w;
#pragma unroll
            for (int m = 0; m < 16; ++m) {
                if (!(m & d)) {
                    float a0r = re[m],     a0i = im[m];
                    float a1r = re[m + d], a1i = im[m + d];
                    re[m]     = c[w] * a0r + s[w] * a1i;
                    im[m]     = c[w] * a0i - s[w] * a1r;
                    re[m + d] = s[w] * a0i + c[w] * a1r;
                    im[m + d] = -s[w] * a0r + c[w] * a1i;
                }
            }
        }
        // CNOT chain (0,1),(1,2),(2,3): swap target where control=1
#pragma unroll
        for (int q = 0; q < 3; ++q) {
            const int dc = 8 >> q;
            const int dt = 8 >> (q + 1);
#pragma unroll
            for (int m = 0; m < 16; ++m) {
                if ((m & dc) && !(m & dt)) {
                    float tr = re[m]; re[m] = re[m + dt]; re[m + dt] = tr;
                    float ti = im[m]; im[m] = im[m + dt]; im[m + dt] = ti;
                }
            }
        }
    }

    float* mre = (tid < 16) ? (ws + 0)   : (ws + 512);
    float* mim = (tid < 16) ? (ws + 256) : (ws + 768);
#pragma unroll
    for (int m = 0; m < 16; ++m) {
        mre[m * 16 + col] = re[m];
        mim[m * 16 + col] = im[m];
    }
}

// ---------------------------------------------------------------------------
// Main kernel: each wave32 processes TILES_PER_WAVE consecutive 16-element
// tiles; batch-uniform A operands (U/V slices) are loaded once per wave.
// Single shfl_xor(16) set per tile: wire-0 rotation is fused into the
// B-operand build (own + partner values derived from one shuffle).
// ---------------------------------------------------------------------------
__global__ void __launch_bounds__(256)
qkm_main(const float* __restrict__ x, const float* __restrict__ y,
         const float* __restrict__ ws, float* __restrict__ out,
         int ntiles, int nelem) {
    const int lane = threadIdx.x & 31;
    const int wave = threadIdx.x >> 5;
    const int wgid = blockIdx.x * (blockDim.x >> 5) + wave;   // wave-uniform
    const int tile0 = wgid * TILES_PER_WAVE;
    if (tile0 >= ntiles) return;                 // wave-uniform: EXEC stays full

    const int n    = lane & 15;                  // column (batch element in tile)
    const int half = lane >> 4;                  // lane half (K/M split per ISA)

    const float* Ure = ws + 0;
    const float* Uim = ws + 256;
    const float* Vre = ws + 512;
    const float* Vim = ws + 768;

    // ---- A operands: rows of U/V in ISA 16x4 fp32 A-layout (hoisted) ----
    const int aoff = n * 16 + 2 * half;
    v2f aUre[4], aUim[4], aVre[4], aVim[4];
#pragma unroll
    for (int cch = 0; cch < 4; ++cch) {
        aUre[cch] = *(const v2f*)(Ure + aoff + 4 * cch);
        aUim[cch] = *(const v2f*)(Uim + aoff + 4 * cch);
        aVre[cch] = *(const v2f*)(Vre + aoff + 4 * cch);
        aVim[cch] = *(const v2f*)(Vim + aoff + 4 * cch);
    }

#pragma unroll 1
    for (int t = 0; t < TILES_PER_WAVE; ++t) {
        const int tile = tile0 + t;
        if (tile >= ntiles) break;               // wave-uniform
        int elem = tile * 16 + n;
        const int valid = (elem < nelem);
        if (elem >= nelem) elem = nelem - 1;     // clamp reads (no-op here)

        // ---- s(x): real separable encoded state, directly in B layout ----
        const v4f xv = *(const v4f*)(x + elem * 4);
        float xc[4], xs[4];
#pragma unroll
        for (int i = 0; i < 4; ++i) {
            float a = 0.5f * xv[i];
            xc[i] = __cosf(a);
            xs[i] = __sinf(a);
        }
        float p01[4] = { xc[0]*xc[1], xc[0]*xs[1], xs[0]*xc[1], xs[0]*xs[1] };
        float p23[4] = { xc[2]*xc[3], xc[2]*xs[3], xs[2]*xc[3], xs[2]*xs[3] };
        // S[m] = p01[m>>2] * p23[m&3]; lane needs K = 4c+2h, 4c+2h+1
        // => bS[c] = ( p01[c]*p23[2h], p01[c]*p23[2h+1] ): only 2 half-selects
        const float q0 = half ? p23[2] : p23[0];
        const float q1 = half ? p23[3] : p23[1];
        v2f bS[4];
#pragma unroll
        for (int cch = 0; cch < 4; ++cch) {
            v2f tt;
            tt.x = p01[cch] * q0;
            tt.y = p01[cch] * q1;
            bS[cch] = tt;
        }

        // ---- GEMM1: T = U * S  (S real) ----
        v8f Tre = {}, Tim = {};
#pragma unroll
        for (int cch = 0; cch < 4; ++cch) {
            Tre = WMMA_F32_4(aUre[cch], bS[cch], Tre);
            Tim = WMMA_F32_4(aUim[cch], bS[cch], Tim);
        }

        // ---- RYten(-y): wires 1..3 first (register-local; they commute
        //      with wire 0, which is fused below) ----
        const v4f yv = *(const v4f*)(y + elem * 4);
        float yc[4], ys[4];
#pragma unroll
        for (int i = 0; i < 4; ++i) {
            float a = 0.5f * yv[i];
            yc[i] = __cosf(a);
            ys[i] = __sinf(a);
        }
        float Qr[8], Qi[8];
#pragma unroll
        for (int r = 0; r < 8; ++r) { Qr[r] = Tre[r]; Qi[r] = Tim[r]; }
#pragma unroll
        for (int w = 1; w < 4; ++w) {
            const int d = 8 >> w;
#pragma unroll
            for (int r = 0; r < 8; ++r) {
                if (!(r & d)) {
                    float ar = Qr[r], ai = Qi[r];
                    float br = Qr[r + d], bi = Qi[r + d];
                    Qr[r]     = yc[w] * ar + ys[w] * br;
                    Qi[r]     = yc[w] * ai + ys[w] * bi;
                    Qr[r + d] = -ys[w] * ar + yc[w] * br;
                    Qi[r + d] = -ys[w] * ai + yc[w] * bi;
                }
            }
        }

        // ---- one shuffle set; fuse wire-0 rotation into own+partner R ----
        // own:     R[r]     = c0*Q[r] + sgn0*q[r]
        // partner: Rpar[r]  = c0*q[r] - sgn0*Q[r]   (partner sign = -sgn0)
        const float sgn0 = half ? -ys[0] : ys[0];
        float Rr[8], Ri[8], Pr[8], Pi[8];
#pragma unroll
        for (int r = 0; r < 8; ++r) {
            float qrp = __shfl_xor(Qr[r], 16, 32);
            float qip = __shfl_xor(Qi[r], 16, 32);
            Rr[r] = yc[0] * Qr[r] + sgn0 * qrp;
            Ri[r] = yc[0] * Qi[r] + sgn0 * qip;
            Pr[r] = yc[0] * qrp - sgn0 * Qr[r];
            Pi[r] = yc[0] * qip - sgn0 * Qi[r];
        }

        // ---- build B operands for GEMM2 (ISA 4x16 B-layout) ----
        v2f bRr[4], bRi[4];
        {
            v2f tt;
            // chunk0 (K=0..3):  low:(M0,M1 own)      high:(M2,M3 partner)
            tt.x = half ? Pr[2] : Rr[0]; tt.y = half ? Pr[3] : Rr[1]; bRr[0] = tt;
            tt.x = half ? Pi[2] : Ri[0]; tt.y = half ? Pi[3] : Ri[1]; bRi[0] = tt;
            // chunk1 (K=4..7):  low:(M4,M5 own)      high:(M6,M7 partner)
            tt.x = half ? Pr[6] : Rr[4]; tt.y = half ? Pr[7] : Rr[5]; bRr[1] = tt;
            tt.x = half ? Pi[6] : Ri[4]; tt.y = half ? Pi[7] : Ri[5]; bRi[1] = tt;
            // chunk2 (K=8..11): low:(M8,M9 partner)  high:(M10,M11 own)
            tt.x = half ? Rr[2] : Pr[0]; tt.y = half ? Rr[3] : Pr[1]; bRr[2] = tt;
            tt.x = half ? Ri[2] : Pi[0]; tt.y = half ? Ri[3] : Pi[1]; bRi[2] = tt;
            // chunk3 (K=12..15): low:(M12,M13 partner) high:(M14,M15 own)
            tt.x = half ? Rr[6] : Pr[4]; tt.y = half ? Rr[7] : Pr[5]; bRr[3] = tt;
            tt.x = half ? Ri[6] : Pi[4]; tt.y = half ? Ri[7] : Pi[5]; bRi[3] = tt;
        }

        // ---- GEMM2: W = V * R  (complex x complex) ----
        v8f P1 = {}, P2 = {}, Wim = {};
#pragma unroll
        for (int cch = 0; cch < 4; ++cch) {
            P1  = WMMA_F32_4(aVre[cch], bRr[cch], P1);   // Vre*Rre
            P2  = WMMA_F32_4(aVim[cch], bRi[cch], P2);   // Vim*Rim
            Wim = WMMA_F32_4(aVre[cch], bRi[cch], Wim);  // Vre*Rim
            Wim = WMMA_F32_4(aVim[cch], bRr[cch], Wim);  // + Vim*Rre
        }
        v8f Wre = P1 - P2;

        // ---- <Z0> = sum_{M<8}|w|^2 - sum_{M>=8}|w|^2 ; out = |ev| ----
        float local = 0.0f;
#pragma unroll
        for (int r = 0; r < 8; ++r)
            local += Wre[r] * Wre[r] + Wim[r] * Wim[r];
        float other = __shfl_xor(local, 16, 32);
        if (half == 0 && valid)
            out[elem] = fabsf(local - other);
    }
}

// ---------------------------------------------------------------------------
extern "C" void kernel_launch(void* const* d_in, const int* in_sizes, int n_in,
                              void* d_out, int out_size, void* d_ws, size_t ws_size,
                              hipStream_t stream) {
    const float* x      = (const float*)d_in[0];   // (BATCH, 4) f32
    const float* y      = (const float*)d_in[1];   // (BATCH, 4) f32
    const float* params = (const float*)d_in[2];   // (4,) f32
    float* out = (float*)d_out;                    // (BATCH,) f32
    float* ws  = (float*)d_ws;                     // 4KB used

    qkm_build_unitaries<<<1, 32, 0, stream>>>(params, ws);

    const int ntiles = (out_size + 15) / 16;                   // 12500
    const int nchunks = (ntiles + TILES_PER_WAVE - 1) / TILES_PER_WAVE;
    const int wavesPerBlock = 8;                                // 256 threads
    const int blocks = (nchunks + wavesPerBlock - 1) / wavesPerBlock;
    qkm_main<<<blocks, 32 * wavesPerBlock, 0, stream>>>(x, y, ws, out,
                                                        ntiles, out_size);
}